// VCN_88484916232248
// MI455X (gfx1250) — compile-verified
//
#include <hip/hip_runtime.h>
#include <hip/hip_bf16.h>

// ---------------------------------------------------------------------------
// Problem shapes (from reference):
//   x_cost : [64, 9, 9, 64, 64] f32   (b=64, u=9, v=9, h=64, w=64)
//   feat   : [4, 128, 64, 64]  f32
//   out    : [4, 162, 64, 64]  f32  = concat(warped[128], flow_mean[2], ent_r[32])
// MD=4, FAC=1, WSIZE=3, FDIMB=16
// ---------------------------------------------------------------------------

typedef __attribute__((ext_vector_type(2))) float v2f;
typedef __attribute__((ext_vector_type(8))) float v8f;

#define HW   4096      // 64*64
#define NB   64        // cost-volume batch
#define NUV  81        // 9*9
#define BOUT 4         // feat batch
#define COUT 162

// ---------------------------------------------------------------------------
// Kernel 1: per-pixel soft-WTA.
// Two passes over the 81 (u,v) planes (stride 4096 floats = 16KB):
//   pass 1: max + argmax
//   pass 2: Zg,Sg (global softmax stats), Zl,Sl,Fx,Fy (windowed stats)
// Entropy via  -sum p log p = log Z - S/Z  with S = sum (x-m) e^{x-m}
// (clip(p,1e-9,..) in the reference only perturbs terms of order 2e-8).
// Writes per-b flow to d_ws, and both entropies straight into out[:,130:162].
// ---------------------------------------------------------------------------
__global__ void k_softwta(const float* __restrict__ xc,
                          float* __restrict__ flowb,   // [64][2][4096]
                          float* __restrict__ out)
{
    const int t  = blockIdx.x * blockDim.x + threadIdx.x;   // 262144 threads
    const int hw = t & (HW - 1);
    const int b  = t >> 12;

    const float* xp = xc + (size_t)b * NUV * HW + hw;

    // ---- pass 1: max / argmax (first occurrence, like jnp.argmax) ----
    float m = -3.402823e38f;
    int   am = 0;
#pragma unroll 9
    for (int i = 0; i < NUV; ++i) {
        // software prefetch a few 16KB-strided planes ahead (gfx1250 global_prefetch_b8)
        int ip = (i + 24 < NUV) ? i + 24 : NUV - 1;
        __builtin_prefetch(xp + (size_t)ip * HW, 0, 0);
        float v = xp[(size_t)i * HW];
        if (v > m) { m = v; am = i; }
    }
    const int ui = am / 9;
    const int vi = am - ui * 9;

    // ---- pass 2: softmax statistics (planes are L2-resident now) ----
    float Zg = 0.f, Sg = 0.f, Zl = 0.f, Sl = 0.f, Fx = 0.f, Fy = 0.f;
#pragma unroll
    for (int u = 0; u < 9; ++u) {
        const int du = u - ui;
        const float uok = (du <= 3 && du >= -3) ? 1.f : 0.f;
        const float fxw = (float)(u - 4);
#pragma unroll
        for (int v = 0; v < 9; ++v) {
            const int dv = v - vi;
            const float wmask = ((dv <= 3 && dv >= -3) ? 1.f : 0.f) * uok;
            const float fyw = (float)(v - 4);
            float x = xp[(size_t)(u * 9 + v) * HW];
            float d = x - m;                 // <= 0, max is inside the window
            float e = __expf(d);
            Zg += e;           Sg += d * e;
            Zl += wmask * e;   Sl += wmask * d * e;
            Fx += wmask * e * fxw;
            Fy += wmask * e * fyw;
        }
    }
    const float invZl = 1.f / Zl;
    const float invZg = 1.f / Zg;
    const float flow_x = Fx * invZl;
    const float flow_y = Fy * invZl;
    const float lent = (__logf(Zl) - Sl * invZl) * (1.0f / 3.8918202981106265f); // /log(49)
    const float gent = (__logf(Zg) - Sg * invZg) * (1.0f / 4.394449154672439f);  // /log(81)

    // per-b flow to workspace (consumed by the WMMA reduction kernel)
    flowb[(size_t)(b * 2 + 0) * HW + hw] = flow_x;
    flowb[(size_t)(b * 2 + 1) * HW + hw] = flow_y;

    // ent_r: [64,2,h,w] flat == [4,32,h,w] flat -> out channels 130..161
    const int B  = b >> 4;
    const int bs = b & 15;
    float* ob = out + ((size_t)B * COUT + 130 + bs * 2) * HW + hw;
    ob[0]  = lent;
    ob[HW] = gent;
}

// ---------------------------------------------------------------------------
// Kernel 2: flow_mean = (1/16) * sum_{k=0..15} flow[B*16+k]  via WMMA.
// D = A x B + C with A = ones(16x4) f32  =>  D[m,n] = sum_k B[k,n].
// Chain four V_WMMA_F32_16X16X4_F32 (K=4 each) to reduce 16 batches exactly
// in f32. Each wave handles 16 pixels for one (B, component).
// B-matrix f32 4x16 lane layout (mirror of documented A 16x4 layout):
//   VGPR0: lanes 0-15 -> (K=0, N=lane), lanes 16-31 -> (K=2, N=lane-16)
//   VGPR1: lanes 0-15 -> (K=1, N=lane), lanes 16-31 -> (K=3, N=lane-16)
// With all-ones A every row of D equals the column sums, so lane l (<16)
// reads the result for pixel l from c[0]. EXEC is all-ones: exact grid,
// no divergence before the WMMA.
// ---------------------------------------------------------------------------
__global__ void k_flowmean_wmma(const float* __restrict__ flowb, // [64][2][4096]
                                float* __restrict__ fm,          // [4][2][4096]
                                float* __restrict__ out)
{
    const int t    = blockIdx.x * blockDim.x + threadIdx.x; // 65536 threads
    const int wid  = t >> 5;                                // 2048 waves
    const int lane = t & 31;
    const int l    = lane & 15;
    const int half = lane >> 4;            // 0: K=0/1, 1: K=2/3

    const int pg   = wid & 255;            // 256 groups of 16 pixels
    const int comp = (wid >> 8) & 1;       // flow component
    const int Bb   = wid >> 9;             // output batch 0..3
    const int pix  = pg * 16 + l;

    const v2f am = {1.0f, 1.0f};           // ones A-matrix (layout-invariant)
    v8f c = {};
#pragma unroll
    for (int k0 = 0; k0 < 16; k0 += 4) {
        const int bb = Bb * 16 + k0 + 2 * half;
        v2f bm;
        bm.x = flowb[(size_t)(bb * 2 + comp) * HW + pix];         // K = k0+2h
        bm.y = flowb[(size_t)((bb + 1) * 2 + comp) * HW + pix];   // K = k0+2h+1
        c = __builtin_amdgcn_wmma_f32_16x16x4_f32(
                /*neg_a=*/false, am, /*neg_b=*/false, bm,
                /*c_mod=*/(short)0, c, /*reuse_a=*/false, /*reuse_b=*/false);
    }

    if (half == 0) {                       // lanes 16-31 hold duplicate rows
        const float mean = c[0] * 0.0625f; // /16
        fm[(size_t)(Bb * 2 + comp) * HW + pix] = mean;
        out[((size_t)Bb * COUT + 128 + comp) * HW + pix] = mean;   // channels 128/129
    }
}

// ---------------------------------------------------------------------------
// Kernel 3: bilinear warp (align_corners=True, zeros padding, in-bounds mask).
// Thread = (B, channel-group of 8, pixel); weights computed once per pixel
// group, corner validity folded into the weights, reused across 8 channels.
// ---------------------------------------------------------------------------
__global__ void k_warp(const float* __restrict__ feat,
                       const float* __restrict__ fm,
                       float* __restrict__ out)
{
    const int t  = blockIdx.x * blockDim.x + threadIdx.x; // 262144 threads
    const int hw = t & (HW - 1);
    const int cg = (t >> 12) & 15;         // 16 groups x 8 channels
    const int Bb = t >> 16;
    const int h  = hw >> 6;
    const int w  = hw & 63;

    const float fx = fm[(size_t)(Bb * 2 + 0) * HW + hw];
    const float fy = fm[(size_t)(Bb * 2 + 1) * HW + hw];
    const float gx = (float)w + fx;
    const float gy = (float)h + fy;
    // |2g/(S-1) - 1| < 1  <=>  0 < g < S-1 (strict), S-1 = 63
    const float inb = (gx > 0.f && gx < 63.f && gy > 0.f && gy < 63.f) ? 1.f : 0.f;

    const float fx0 = floorf(gx), fy0 = floorf(gy);
    const int x0 = (int)fx0, y0 = (int)fy0;
    const int x1 = x0 + 1,   y1 = y0 + 1;
    const float wx = gx - fx0, wy = gy - fy0;

    const float vx0 = (x0 >= 0 && x0 < 64) ? 1.f : 0.f;
    const float vx1 = (x1 >= 0 && x1 < 64) ? 1.f : 0.f;
    const float vy0 = (y0 >= 0 && y0 < 64) ? 1.f : 0.f;
    const float vy1 = (y1 >= 0 && y1 < 64) ? 1.f : 0.f;
    const int cx0 = min(max(x0, 0), 63), cx1 = min(max(x1, 0), 63);
    const int cy0 = min(max(y0, 0), 63), cy1 = min(max(y1, 0), 63);

    const float w00 = (1.f - wy) * (1.f - wx) * vy0 * vx0;
    const float w01 = (1.f - wy) * wx         * vy0 * vx1;
    const float w10 = wy         * (1.f - wx) * vy1 * vx0;
    const float w11 = wy         * wx         * vy1 * vx1;

    const int i00 = cy0 * 64 + cx0, i01 = cy0 * 64 + cx1;
    const int i10 = cy1 * 64 + cx0, i11 = cy1 * 64 + cx1;

    const float* fb = feat + ((size_t)Bb * 128 + cg * 8) * HW;
    float* op = out + ((size_t)Bb * COUT + cg * 8) * HW + hw;
#pragma unroll
    for (int cch = 0; cch < 8; ++cch) {
        const float* fc = fb + (size_t)cch * HW;
        float r = w00 * fc[i00] + w01 * fc[i01] + w10 * fc[i10] + w11 * fc[i11];
        op[(size_t)cch * HW] = r * inb;
    }
}

// ---------------------------------------------------------------------------
extern "C" void kernel_launch(void* const* d_in, const int* in_sizes, int n_in,
                              void* d_out, int out_size, void* d_ws, size_t ws_size,
                              hipStream_t stream)
{
    (void)in_sizes; (void)n_in; (void)out_size; (void)ws_size;

    const float* x_cost = (const float*)d_in[0];   // [64,9,9,64,64]
    const float* feat   = (const float*)d_in[1];   // [4,128,64,64]
    float* out = (float*)d_out;                    // [4,162,64,64]

    // workspace: per-b flow [64][2][4096] (2 MB) then flow_mean [4][2][4096]
    float* flowb = (float*)d_ws;
    float* fm    = flowb + (size_t)NB * 2 * HW;

    // 1) streaming soft-WTA reduction: 262144 pixels
    k_softwta<<<1024, 256, 0, stream>>>(x_cost, flowb, out);
    // 2) 16-way batch mean via chained v_wmma_f32_16x16x4_f32: 2048 waves
    k_flowmean_wmma<<<256, 256, 0, stream>>>(flowb, fm, out);
    // 3) bilinear warp, 8 channels per thread
    k_warp<<<1024, 256, 0, stream>>>(feat, fm, out);
}